// GraphSAGE_7524782702739
// MI455X (gfx1250) — compile-verified
//
#include <hip/hip_runtime.h>

typedef __attribute__((ext_vector_type(2))) float v2f;
typedef __attribute__((ext_vector_type(8))) float v8f;

#define N_NODES 100000
#define IN_C 128
#define HID 16
#define OUT_C 128

// ---------------------------------------------------------------------------
// Zero a float buffer (agg / deg re-init each call; ws is not re-poisoned).
// ---------------------------------------------------------------------------
__global__ void sage_zero_kernel(float* __restrict__ buf, int n) {
    int gid = blockIdx.x * blockDim.x + threadIdx.x;
    if (gid < n) buf[gid] = 0.0f;
}

// ---------------------------------------------------------------------------
// Layer-1 projections BEFORE aggregation (mean is linear):
//   p  = x @ W1   [N,16]
//   r1 = x @ R1   [N,16]
// One wave handles one 16-row tile; K-loop of 32 steps of V_WMMA_F32_16X16X4_F32.
// A-frag layout (ISA 7.12.2, 32-bit A 16x4): lane<16 -> M=lane, K=k0+0/1;
// lane>=16 -> M=lane-16, K=k0+2/3. B-frag symmetric. C/D: VGPR i -> M=i+8*half.
// ---------------------------------------------------------------------------
__global__ void sage_proj1_kernel(const float* __restrict__ x,
                                  const float* __restrict__ W1,
                                  const float* __restrict__ R1,
                                  float* __restrict__ p,
                                  float* __restrict__ r1) {
    const int wave = (blockIdx.x * blockDim.x + threadIdx.x) >> 5;
    const int lane = threadIdx.x & 31;
    const int ntiles = N_NODES / 16;           // 6250, exact
    if (wave >= ntiles) return;                // wave-uniform: EXEC stays all-1s
    const int row0 = wave * 16;
    const int half = lane >> 4;
    const int l16  = lane & 15;

    v8f cp = {};   // p accumulator
    v8f cr = {};   // r1 accumulator

    const float* xrow = x + (long long)(row0 + l16) * IN_C + 2 * half;
    #pragma unroll 4
    for (int k0 = 0; k0 < IN_C; k0 += 4) {
        v2f a;
        a.x = xrow[k0];
        a.y = xrow[k0 + 1];
        const int kb = k0 + 2 * half;
        v2f bw, br;
        bw.x = W1[(kb)     * HID + l16];
        bw.y = W1[(kb + 1) * HID + l16];
        br.x = R1[(kb)     * HID + l16];
        br.y = R1[(kb + 1) * HID + l16];
        cp = __builtin_amdgcn_wmma_f32_16x16x4_f32(false, a, false, bw,
                                                   (short)0, cp, false, false);
        cr = __builtin_amdgcn_wmma_f32_16x16x4_f32(false, a, false, br,
                                                   (short)0, cr, false, false);
    }

    #pragma unroll
    for (int i = 0; i < 8; ++i) {
        const int m = row0 + i + 8 * half;
        p [m * HID + l16] = cp[i];
        r1[m * HID + l16] = cr[i];
    }
}

// ---------------------------------------------------------------------------
// Edge scatter: lane-per-(edge,feature). 16 consecutive lanes cover one edge's
// 16 features -> coalesced gather of feat[src], L2-resident float atomics into
// agg[dst]. Optionally counts in-degree (pass 1 only).
// ---------------------------------------------------------------------------
__global__ void sage_edge_agg_kernel(const long long* __restrict__ src,
                                     const long long* __restrict__ dst,
                                     const float* __restrict__ feat,
                                     float* __restrict__ agg,
                                     float* __restrict__ deg,
                                     int n_edges, int count_deg) {
    long long gid = (long long)blockIdx.x * blockDim.x + threadIdx.x;
    if (gid >= (long long)n_edges * HID) return;
    const int e = (int)(gid >> 4);
    const int f = (int)(gid & 15);
    const int s = (int)src[e];
    const int d = (int)dst[e];
    atomicAdd(&agg[d * HID + f], feat[s * HID + f]);
    if (count_deg && f == 0) atomicAdd(&deg[d], 1.0f);
}

// ---------------------------------------------------------------------------
// h = relu(agg/max(deg,1) + b1 + r1)
// ---------------------------------------------------------------------------
__global__ void sage_finalize1_kernel(const float* __restrict__ agg,
                                      const float* __restrict__ deg,
                                      const float* __restrict__ r1,
                                      const float* __restrict__ b1,
                                      float* __restrict__ h) {
    int gid = blockIdx.x * blockDim.x + threadIdx.x;
    if (gid >= N_NODES * HID) return;
    const int n = gid >> 4;
    const int f = gid & 15;
    const float dv = fmaxf(deg[n], 1.0f);
    const float v = agg[gid] / dv + b1[f] + r1[gid];
    h[gid] = fmaxf(v, 0.0f);
}

// ---------------------------------------------------------------------------
// Layer 2: out = (agg2/deg) @ W2 + b2 + h @ R2   ([N,16]@[16,128] -> [N,128])
// One wave per 16-row tile: A-frags (mean and h, K=16 -> 4 k-steps) are loaded
// once and reused across 8 N-tiles; 8 WMMAs per N-tile accumulate both paths.
// ---------------------------------------------------------------------------
__global__ void sage_layer2_kernel(const float* __restrict__ h,
                                   const float* __restrict__ agg2,
                                   const float* __restrict__ deg,
                                   const float* __restrict__ W2,
                                   const float* __restrict__ R2,
                                   const float* __restrict__ b2,
                                   float* __restrict__ out) {
    const int wave = (blockIdx.x * blockDim.x + threadIdx.x) >> 5;
    const int lane = threadIdx.x & 31;
    const int ntiles = N_NODES / 16;
    if (wave >= ntiles) return;                // wave-uniform exit
    const int row0 = wave * 16;
    const int half = lane >> 4;
    const int l16  = lane & 15;

    const int mrow = row0 + l16;               // this lane's A-matrix row
    const float inv = 1.0f / fmaxf(deg[mrow], 1.0f);

    v2f am[4], ah[4];
    #pragma unroll
    for (int j = 0; j < 4; ++j) {
        const int k = 4 * j + 2 * half;
        am[j].x = agg2[mrow * HID + k]     * inv;
        am[j].y = agg2[mrow * HID + k + 1] * inv;
        ah[j].x = h[mrow * HID + k];
        ah[j].y = h[mrow * HID + k + 1];
    }

    #pragma unroll
    for (int nt = 0; nt < 8; ++nt) {
        const int ncol = nt * 16 + l16;
        v8f c = {};
        #pragma unroll
        for (int j = 0; j < 4; ++j) {
            const int k = 4 * j + 2 * half;
            v2f bw, br;
            bw.x = W2[(k)     * OUT_C + ncol];
            bw.y = W2[(k + 1) * OUT_C + ncol];
            c = __builtin_amdgcn_wmma_f32_16x16x4_f32(false, am[j], false, bw,
                                                      (short)0, c, false, false);
            br.x = R2[(k)     * OUT_C + ncol];
            br.y = R2[(k + 1) * OUT_C + ncol];
            c = __builtin_amdgcn_wmma_f32_16x16x4_f32(false, ah[j], false, br,
                                                      (short)0, c, false, false);
        }
        const float bias = b2[ncol];
        #pragma unroll
        for (int i = 0; i < 8; ++i) {
            const int m = row0 + i + 8 * half;
            out[(long long)m * OUT_C + ncol] = c[i] + bias;
        }
    }
}

// ---------------------------------------------------------------------------
extern "C" void kernel_launch(void* const* d_in, const int* in_sizes, int n_in,
                              void* d_out, int out_size, void* d_ws, size_t ws_size,
                              hipStream_t stream) {
    const float*     x  = (const float*)d_in[0];
    const long long* ei = (const long long*)d_in[1];   // int64 per reference
    const float*     W1 = (const float*)d_in[2];
    const float*     b1 = (const float*)d_in[3];
    const float*     R1 = (const float*)d_in[4];
    const float*     W2 = (const float*)d_in[5];
    const float*     b2 = (const float*)d_in[6];
    const float*     R2 = (const float*)d_in[7];
    float* out = (float*)d_out;

    const int n_edges = in_sizes[1] / 2;
    const long long* src = ei;
    const long long* dst = ei + n_edges;

    // Workspace layout (floats): p | r1 | h | agg | deg  (~26 MB)
    float* ws  = (float*)d_ws;
    float* p   = ws;
    float* r1  = p   + (size_t)N_NODES * HID;
    float* h   = r1  + (size_t)N_NODES * HID;
    float* agg = h   + (size_t)N_NODES * HID;
    float* deg = agg + (size_t)N_NODES * HID;   // agg and deg are contiguous

    const int ntiles = N_NODES / 16;            // 6250
    const int gemm_blocks = (ntiles + 7) / 8;   // 8 waves (256 thr) per block

    // --- Layer 1 ---
    {   // zero agg (N*16) + deg (N) in one shot (contiguous)
        int n = N_NODES * HID + N_NODES;
        sage_zero_kernel<<<(n + 255) / 256, 256, 0, stream>>>(agg, n);
    }
    sage_proj1_kernel<<<gemm_blocks, 256, 0, stream>>>(x, W1, R1, p, r1);
    {
        long long work = (long long)n_edges * HID;
        int blocks = (int)((work + 255) / 256);
        sage_edge_agg_kernel<<<blocks, 256, 0, stream>>>(src, dst, p, agg, deg,
                                                         n_edges, 1);
    }
    {
        int n = N_NODES * HID;
        sage_finalize1_kernel<<<(n + 255) / 256, 256, 0, stream>>>(agg, deg, r1, b1, h);
    }

    // --- Layer 2 ---
    {
        int n = N_NODES * HID;
        sage_zero_kernel<<<(n + 255) / 256, 256, 0, stream>>>(agg, n);
    }
    {
        long long work = (long long)n_edges * HID;
        int blocks = (int)((work + 255) / 256);
        sage_edge_agg_kernel<<<blocks, 256, 0, stream>>>(src, dst, h, agg, deg,
                                                         n_edges, 0);
    }
    sage_layer2_kernel<<<gemm_blocks, 256, 0, stream>>>(h, agg, deg, W2, R2, b2, out);
}